// Shap_part_2_51977694216602
// MI455X (gfx1250) — compile-verified
//
#include <hip/hip_runtime.h>

#define M_FEAT 32
#define N_COAL 4096
#define N_BG   256
#define N_INST 64

typedef __attribute__((ext_vector_type(2))) float v2f;
typedef __attribute__((ext_vector_type(8))) float v8f;

// workspace layout (float elements)
#define WS_Y    0                          // y_mean: B*C
#define WS_PI   (N_INST * N_COAL)          // pi_c:   C
#define WS_AINV (WS_PI + N_COAL)           // Ainv:   32*32
#define WS_T    (WS_AINV + M_FEAT*M_FEAT)  // t:      64*32

// ---------------------------------------------------------------------------
// K1: bandwidth kernel. One wave32 per (b,c) row of 256 contiguous floats.
// Each lane: two coalesced float4 loads (covers 128 + 128 floats), then a
// 5-step __shfl_xor wave reduction. ~268 MB streamed -> HBM-bound.
// ---------------------------------------------------------------------------
__global__ void k1_row_mean(const float* __restrict__ preds,
                            float* __restrict__ y_mean) {
  const int wave = threadIdx.x >> 5;
  const int lane = threadIdx.x & 31;
  const size_t r = (size_t)blockIdx.x * 8 + (size_t)wave;   // (b*C + c)
  const float4* base = reinterpret_cast<const float4*>(preds + r * (size_t)N_BG);
  float4 a = base[lane];
  float4 b = base[lane + 32];
  float s = (a.x + a.y) + (a.z + a.w) + (b.x + b.y) + (b.z + b.w);
#pragma unroll
  for (int off = 16; off > 0; off >>= 1) s += __shfl_xor(s, off, 32);
  if (lane == 0) y_mean[r] = s * (1.0f / (float)N_BG);
}

// ---------------------------------------------------------------------------
// K2: pi weights, A = cv^T diag(pi) cv, and A^-1 via f64 Gauss-Jordan (LDS).
// Single 1024-thread block; thread (i,j) owns A[i][j].
// ---------------------------------------------------------------------------
__device__ __forceinline__ double buggy_fact(int n) {
  return (n < 2) ? 1.0 : (double)n * (double)(n + 1) * 0.5;
}
__device__ __forceinline__ float pi_of_k(int k) {
  if (k <= 0 || k >= M_FEAT) return 1.0e9f;
  double comb = buggy_fact(M_FEAT) / (buggy_fact(k) * buggy_fact(M_FEAT - k));
  return (float)((double)(M_FEAT - 1) / (comb * (double)k * (double)(M_FEAT - k)));
}

__global__ void k2_setup(const float* __restrict__ cv, float* __restrict__ ws) {
  float* pi_c = ws + WS_PI;
  float* Ainv = ws + WS_AINV;
  const int tid = threadIdx.x;            // 1024 threads
  // phase 1: per-coalition weight
  for (int c = tid; c < N_COAL; c += 1024) {
    int k = 0;
    for (int i = 0; i < M_FEAT; ++i) k += (cv[c * M_FEAT + i] > 0.0f) ? 1 : 0;
    float p = pi_of_k(k);
    if (c == 0 || c == N_COAL - 1) p = 1.0e9f;
    pi_c[c] = p;
  }
  __syncthreads();
  // phase 2: A[i][j] = sum_c pi_c * cv[c][i] * cv[c][j]  (f32 like reference)
  const int i = tid >> 5, j = tid & 31;
  float a = 0.0f;
  for (int c = 0; c < N_COAL; ++c)
    a = fmaf(pi_c[c] * cv[c * M_FEAT + i], cv[c * M_FEAT + j], a);
  // phase 3: Gauss-Jordan with partial pivoting, double precision in LDS
  __shared__ double aug[M_FEAT][2 * M_FEAT];   // 16 KB
  __shared__ int s_piv;
  aug[i][j] = (double)a;
  aug[i][j + 32] = (i == j) ? 1.0 : 0.0;
  __syncthreads();
  for (int k = 0; k < M_FEAT; ++k) {
    if (tid == 0) {
      int p = k; double best = fabs(aug[k][k]);
      for (int rr = k + 1; rr < M_FEAT; ++rr) {
        double v = fabs(aug[rr][k]);
        if (v > best) { best = v; p = rr; }
      }
      s_piv = p;
    }
    __syncthreads();
    const int p = s_piv;
    if (p != k && tid < 64) {
      double tmp = aug[k][tid]; aug[k][tid] = aug[p][tid]; aug[p][tid] = tmp;
    }
    __syncthreads();
    const double pivval = aug[k][k];
    __syncthreads();
    if (tid < 64) aug[k][tid] /= pivval;
    __syncthreads();
    const double f = (i != k) ? aug[i][k] : 0.0;
    __syncthreads();
    if (i != k) {
      aug[i][j]      -= f * aug[k][j];
      aug[i][j + 32] -= f * aug[k][j + 32];
    }
    __syncthreads();
  }
  Ainv[i * M_FEAT + j] = (float)aug[i][j + 32];
}

// ---------------------------------------------------------------------------
// K3: t = (y - phi0) @ W, W[c][j] = pi_c[c]*cv[c][j], via V_WMMA_F32_16X16X4_F32.
// One wave per 16x16 output tile (4 b-tiles x 2 j-tiles = 8 waves, one block).
// ISA layouts (wave32):
//   A 16x4 f32 : lanes 0-15 row=lane,  VGPR0=K0, VGPR1=K1; lanes 16-31: K2,K3
//   B  4x16    : VGPR v, half h -> row K = v + 2h, col = lane%16
//   C/D 16x16  : VGPR v, half h -> row M = v + 8h, col = lane%16
// ---------------------------------------------------------------------------
__global__ void k3_wmma_t(const float* __restrict__ y_mean,
                          const float* __restrict__ pi_c,
                          const float* __restrict__ cv,
                          float* __restrict__ t_out) {
  const int wave = threadIdx.x >> 5;
  const int lane = threadIdx.x & 31;
  const int half = lane >> 4;
  const int l16  = lane & 15;
  const int btile = wave >> 1;            // 0..3  (b rows)
  const int jtile = wave & 1;             // 0..1  (j cols)
  const float phi0 = y_mean[0];
  const int rA = btile * 16 + l16;        // b row fed by this lane
  const int jB = jtile * 16 + l16;        // j col fed by this lane
  const float* yrow = y_mean + (size_t)rA * N_COAL;

  v8f acc = {0.f, 0.f, 0.f, 0.f, 0.f, 0.f, 0.f, 0.f};
  for (int c0 = 0; c0 < N_COAL; c0 += 4) {
    const int kb = c0 + (half << 1);      // this half supplies K = kb, kb+1
    v2f a, b;
    a.x = yrow[kb]     - phi0;
    a.y = yrow[kb + 1] - phi0;
    b.x = pi_c[kb]     * cv[(size_t)kb * M_FEAT + jB];
    b.y = pi_c[kb + 1] * cv[(size_t)(kb + 1) * M_FEAT + jB];
    acc = __builtin_amdgcn_wmma_f32_16x16x4_f32(false, a, false, b,
                                                (short)0, acc, false, false);
  }
#pragma unroll
  for (int v = 0; v < 8; ++v) {
    const int m = btile * 16 + v + 8 * half;   // b
    const int n = jtile * 16 + l16;            // j
    t_out[m * M_FEAT + n] = acc[v];
  }
}

// ---------------------------------------------------------------------------
// K4: shap[b][m] = sum_j Ainv[m][j] * t[b][j]; labels; phi0.
// ---------------------------------------------------------------------------
__global__ void k4_final(const float* __restrict__ preds,
                         const float* __restrict__ ws,
                         float* __restrict__ out) {
  const float* y    = ws + WS_Y;
  const float* Ainv = ws + WS_AINV;
  const float* t    = ws + WS_T;
  const int tid = threadIdx.x;            // 1024 threads
  if (tid < N_INST)
    out[tid] = preds[(size_t)N_INST * N_COAL * N_BG + (size_t)tid];
  for (int o = tid; o < N_INST * M_FEAT; o += 1024) {
    const int b = o >> 5, m = o & 31;
    float s = 0.0f;
#pragma unroll
    for (int j = 0; j < M_FEAT; ++j)
      s = fmaf(Ainv[m * M_FEAT + j], t[b * M_FEAT + j], s);
    out[N_INST + o] = s;                  // o == b*32 + m
  }
  if (tid == 0) out[N_INST + N_INST * M_FEAT] = y[0];   // phi0
}

// ---------------------------------------------------------------------------
extern "C" void kernel_launch(void* const* d_in, const int* in_sizes, int n_in,
                              void* d_out, int out_size, void* d_ws, size_t ws_size,
                              hipStream_t stream) {
  (void)in_sizes; (void)n_in; (void)out_size; (void)ws_size;
  const float* preds = (const float*)d_in[1];
  const float* cv    = (const float*)d_in[3];
  float* ws  = (float*)d_ws;
  float* out = (float*)d_out;

  k1_row_mean<<<(N_INST * N_COAL) / 8, 256, 0, stream>>>(preds, ws + WS_Y);
  k2_setup  <<<1, 1024, 0, stream>>>(cv, ws);
  k3_wmma_t <<<1, 256, 0, stream>>>(ws + WS_Y, ws + WS_PI, cv, ws + WS_T);
  k4_final  <<<1, 1024, 0, stream>>>(preds, ws, out);
}